// SparseVectorQuantizer_1005022347467
// MI455X (gfx1250) — compile-verified
//
#include <hip/hip_runtime.h>

typedef float v2f __attribute__((ext_vector_type(2)));
typedef float v8f __attribute__((ext_vector_type(8)));
typedef int   v4i __attribute__((vector_size(4 * sizeof(int))));

#define N_VOX 65536
#define K_EMB 8192
#define D_DIM 64

#define ROWS_PER_WG 32   // two 16-row WMMA tiles per workgroup
#define KC 64            // codes staged to LDS per outer iteration
#define NT (K_EMB / KC)  // 128 outer iterations
#define LDS_STRIDE 68    // 64 + 4 pad: conflict-free B reads, 16B-aligned rows

#if __has_builtin(__builtin_amdgcn_global_load_async_to_lds_b128) && \
    __has_builtin(__builtin_amdgcn_s_wait_asynccnt)
#define USE_ASYNC_LDS 1
#define AS_GLOBAL __attribute__((address_space(1)))
#define AS_LDS    __attribute__((address_space(3)))
#else
#define USE_ASYNC_LDS 0
#endif

// ---------------- kernel 1: e_sq[c] = sum_d emb[c][d]^2 ----------------
__global__ __launch_bounds__(256) void esq_kernel(const float* __restrict__ emb,
                                                  float* __restrict__ esq) {
  int c = blockIdx.x * 256 + threadIdx.x;
  const float4* row = (const float4*)(emb + (size_t)c * D_DIM);
  float s = 0.f;
#pragma unroll
  for (int i = 0; i < 16; ++i) {
    float4 v = row[i];
    s += v.x * v.x + v.y * v.y + v.z * v.z + v.w * v.w;
  }
  esq[c] = s;
}

// ---------------- kernel 2: fused distance + argmin + gather ----------------
__global__ __launch_bounds__(256) void vq_kernel(const float* __restrict__ z,
                                                 const float* __restrict__ emb,
                                                 const float* __restrict__ esq,
                                                 float* __restrict__ out_q,
                                                 float* __restrict__ out_idx,
                                                 float* __restrict__ blk_loss) {
#if USE_ASYNC_LDS
  __shared__ float lds_emb[2][KC * LDS_STRIDE];
#else
  __shared__ float lds_emb[1][KC * LDS_STRIDE];
#endif
  __shared__ float red_val[4][ROWS_PER_WG];
  __shared__ int   red_idx[4][ROWS_PER_WG];
  __shared__ int   final_idx[ROWS_PER_WG];
  __shared__ float red_loss[256];

  const int tid  = threadIdx.x;
  const int wave = tid >> 5;   // 0..7 (wave32)
  const int lane = tid & 31;
  const int rt   = wave >> 2;  // row-tile 0/1
  const int wc   = wave & 3;   // code-column group 0..3
  const int l16  = lane & 15;
  const int h    = lane >> 4;  // half-wave selector

  const int row_base = blockIdx.x * ROWS_PER_WG;
  const int grow = row_base + rt * 16 + l16;

  // staging assignment: thread -> (code, dim-group) within the KC x 64 tile
  const int sc  = tid >> 2;        // staged code 0..63
  const int sdq = (tid & 3) * 16;  // staged dim offset {0,16,32,48}

  // Preload A fragments (wave32 16x4 fp32 A layout):
  // lane holds row (lane&15); VGPR pair covers dims 4*s + 2*h, +1
  v2f areg[16];
  const float* zrow = z + (size_t)grow * D_DIM;
#pragma unroll
  for (int s = 0; s < 16; ++s)
    areg[s] = *(const v2f*)(zrow + 4 * s + 2 * h);

  float minv[8];
  int   mini[8];
#pragma unroll
  for (int j = 0; j < 8; ++j) { minv[j] = 3.4e38f; mini[j] = 0; }

#if USE_ASYNC_LDS
  // -------- async global->LDS double-buffered pipeline (ASYNCcnt) --------
  {
    // prologue: stage tile 0 into buffer 0
    const float* gsrc = emb + (size_t)sc * D_DIM + sdq;
    float* ldst = &lds_emb[0][sc * LDS_STRIDE + sdq];
#pragma unroll
    for (int q = 0; q < 4; ++q)
      __builtin_amdgcn_global_load_async_to_lds_b128(
          (AS_GLOBAL v4i*)(gsrc + 4 * q),
          (AS_LDS v4i*)(ldst + 4 * q), 0, 0);
  }
  for (int it = 0; it < NT; ++it) {
    const int kb = it * KC;
    __syncthreads();  // B1: buffer (it+1)&1 no longer being read
    if (it + 1 < NT) {
      const float* gsrc = emb + (size_t)(kb + KC + sc) * D_DIM + sdq;
      float* ldst = &lds_emb[(it + 1) & 1][sc * LDS_STRIDE + sdq];
#pragma unroll
      for (int q = 0; q < 4; ++q)
        __builtin_amdgcn_global_load_async_to_lds_b128(
            (AS_GLOBAL v4i*)(gsrc + 4 * q),
            (AS_LDS v4i*)(ldst + 4 * q), 0, 0);
      __builtin_amdgcn_s_wait_asynccnt(4);  // retire current buffer's 4 ops
    } else {
      __builtin_amdgcn_s_wait_asynccnt(0);
    }
    __syncthreads();  // B2: current buffer visible workgroup-wide
    const float* ldsbuf = lds_emb[it & 1];
#else
  // -------- fallback: register-prefetch software pipeline --------
  float4 stage[4];
  {
    const float4* src = (const float4*)(emb + (size_t)sc * D_DIM + sdq);
#pragma unroll
    for (int q = 0; q < 4; ++q) stage[q] = src[q];
  }
  for (int it = 0; it < NT; ++it) {
    const int kb = it * KC;
    __syncthreads();  // B1: previous tile consumers done
    {
      float4* dst = (float4*)(&lds_emb[0][sc * LDS_STRIDE + sdq]);
#pragma unroll
      for (int q = 0; q < 4; ++q) dst[q] = stage[q];
    }
    if (it + 1 < NT) {  // issue next tile's global loads behind the compute
      const float4* src = (const float4*)(emb + (size_t)(kb + KC + sc) * D_DIM + sdq);
#pragma unroll
      for (int q = 0; q < 4; ++q) stage[q] = src[q];
    }
    __syncthreads();  // B2
    const float* ldsbuf = lds_emb[0];
#endif

    // 16 x v_wmma_f32_16x16x4_f32: C[16 rows][16 codes] += A[16x4] * B[4x16]
    v8f acc0 = {};
    v8f acc1 = {};
    const float* bbase = ldsbuf + (wc * 16 + l16) * LDS_STRIDE + 2 * h;
#pragma unroll
    for (int s = 0; s < 16; s += 2) {
      v2f b0 = *(const v2f*)(bbase + 4 * s);
      v2f b1 = *(const v2f*)(bbase + 4 * (s + 1));
      acc0 = __builtin_amdgcn_wmma_f32_16x16x4_f32(false, areg[s],     false, b0,
                                                   (short)0, acc0, false, false);
      acc1 = __builtin_amdgcn_wmma_f32_16x16x4_f32(false, areg[s + 1], false, b1,
                                                   (short)0, acc1, false, false);
    }

    const int   code = kb + wc * 16 + l16;
    const float es   = esq[code];
#pragma unroll
    for (int j = 0; j < 8; ++j) {
      float sc2 = es - 2.0f * (acc0[j] + acc1[j]);
      if (sc2 < minv[j]) { minv[j] = sc2; mini[j] = code; }  // codes ascend -> first-min kept
    }
  }

  // Cross-lane min+argmin butterfly within each 16-lane half (C layout:
  // VGPR j, lanes 0-15 -> row j; lanes 16-31 -> row j+8; column = lane&15)
#pragma unroll
  for (int j = 0; j < 8; ++j) {
    float v = minv[j];
    int   i = mini[j];
#pragma unroll
    for (int m = 1; m <= 8; m <<= 1) {
      float ov = __shfl_xor(v, m, 32);
      int   oi = __shfl_xor(i, m, 32);
      if (ov < v || (ov == v && oi < i)) { v = ov; i = oi; }
    }
    if (l16 == 0) {
      int r = rt * 16 + j + h * 8;
      red_val[wc][r] = v;
      red_idx[wc][r] = i;
    }
  }
  __syncthreads();

  // Final per-row reduce across the 4 code-column waves
  if (tid < ROWS_PER_WG) {
    float v = red_val[0][tid];
    int   i = red_idx[0][tid];
#pragma unroll
    for (int w = 1; w < 4; ++w) {
      float ov = red_val[w][tid];
      int   oi = red_idx[w][tid];
      if (ov < v || (ov == v && oi < i)) { v = ov; i = oi; }
    }
    final_idx[tid] = i;
    out_idx[row_base + tid] = (float)i;
  }
  __syncthreads();

  // Gather quantized rows + deterministic per-block loss partial
  {
    const int r  = tid >> 3;
    const int c0 = (tid & 7) * 8;
    const int qi = final_idx[r];
    const float4* qsrc = (const float4*)(emb + (size_t)qi * D_DIM + c0);
    const float4* zsrc = (const float4*)(z + (size_t)(row_base + r) * D_DIM + c0);
    float4* qdst = (float4*)(out_q + (size_t)(row_base + r) * D_DIM + c0);
    float lsum = 0.f;
#pragma unroll
    for (int q = 0; q < 2; ++q) {
      float4 qv = qsrc[q], zv = zsrc[q];
      float dx = zv.x - qv.x, dy = zv.y - qv.y;
      float dz = zv.z - qv.z, dw = zv.w - qv.w;
      lsum += dx * dx + dy * dy + dz * dz + dw * dw;
      qdst[q] = qv;
    }
    red_loss[tid] = lsum;
  }
  __syncthreads();
  for (int s = 128; s > 0; s >>= 1) {
    if (tid < s) red_loss[tid] += red_loss[tid + s];
    __syncthreads();
  }
  if (tid == 0) blk_loss[blockIdx.x] = red_loss[0];
}

// ---------------- kernel 3: deterministic final loss reduction ----------------
__global__ __launch_bounds__(256) void loss_kernel(const float* __restrict__ blk_loss,
                                                   float* __restrict__ out_loss,
                                                   int nblk) {
  __shared__ float red[256];
  float s = 0.f;
  for (int i = threadIdx.x; i < nblk; i += 256) s += blk_loss[i];
  red[threadIdx.x] = s;
  __syncthreads();
  for (int st = 128; st > 0; st >>= 1) {
    if (threadIdx.x < st) red[threadIdx.x] += red[threadIdx.x + st];
    __syncthreads();
  }
  if (threadIdx.x == 0) {
    float m = red[0] / (float)((size_t)N_VOX * D_DIM);
    out_loss[0] = m;  // vq_loss
    out_loss[1] = m;  // commitment_loss (numerically identical)
  }
}

extern "C" void kernel_launch(void* const* d_in, const int* in_sizes, int n_in,
                              void* d_out, int out_size, void* d_ws, size_t ws_size,
                              hipStream_t stream) {
  const float* z   = (const float*)d_in[0];  // [N, 64]
  const float* emb = (const float*)d_in[1];  // [K, 64]

  float* out      = (float*)d_out;
  float* out_q    = out;                               // [N*64]
  float* out_loss = out + (size_t)N_VOX * D_DIM;       // vq_loss, commitment_loss
  float* out_idx  = out_loss + 2;                      // [N]

  // workspace: [0, 2048) block loss partials; [2048, 2048+K) e_sq
  float* ws       = (float*)d_ws;
  float* blk_loss = ws;
  float* esq      = ws + (N_VOX / ROWS_PER_WG);

  esq_kernel<<<K_EMB / 256, 256, 0, stream>>>(emb, esq);
  vq_kernel<<<N_VOX / ROWS_PER_WG, 256, 0, stream>>>(z, emb, esq, out_q, out_idx, blk_loss);
  loss_kernel<<<1, 256, 0, stream>>>(blk_loss, out_loss, N_VOX / ROWS_PER_WG);
}